// RoIHeads_41721312313796
// MI455X (gfx1250) — compile-verified
//
#include <hip/hip_runtime.h>
#include <hip/hip_bf16.h>
#include <math.h>

// Problem constants (fixed in the reference file)
#define R_    2048
#define C_    81
#define M_    28
#define G_    100
#define H_    512
#define W_    512
#define NPOS_ 512
#define MM_   (M_ * M_)   // 784

typedef float v2f __attribute__((ext_vector_type(2)));
typedef float v8f __attribute__((ext_vector_type(8)));

// ---------------------------------------------------------------------------
// Kernel 1: per-row -log_softmax at the label column. One wave32 per row.
// ---------------------------------------------------------------------------
__global__ __launch_bounds__(256) void cls_rows_kernel(
    const float* __restrict__ logits, const int* __restrict__ label,
    float* __restrict__ cls_rows) {
  int wave = (blockIdx.x * blockDim.x + threadIdx.x) >> 5;
  int lane = threadIdx.x & 31;
  if (wave >= R_) return;
  const float* row = logits + (size_t)wave * C_;

  float v0 = row[lane];                                          // lane < 32 < 81
  float v1 = row[lane + 32];                                     // lane+32 < 81
  float v2 = (lane + 64 < C_) ? row[lane + 64] : -INFINITY;      // lane < 17
  float m = fmaxf(v0, fmaxf(v1, v2));
  #pragma unroll
  for (int off = 16; off; off >>= 1) m = fmaxf(m, __shfl_xor(m, off, 32));

  float s = expf(v0 - m) + expf(v1 - m) + expf(v2 - m);          // exp(-inf)=0
  #pragma unroll
  for (int off = 16; off; off >>= 1) s += __shfl_xor(s, off, 32);

  if (lane == 0) {
    float ll = row[label[wave]];
    cls_rows[wave] = -(ll - m - logf(s));
  }
}

// ---------------------------------------------------------------------------
// Kernel 2: smooth-L1 per positive ROI (sum over its 4 coords).
// ---------------------------------------------------------------------------
__global__ __launch_bounds__(256) void box_rows_kernel(
    const float* __restrict__ breg, const float* __restrict__ tgt,
    const int* __restrict__ label, float* __restrict__ box_rows) {
  int r = blockIdx.x * blockDim.x + threadIdx.x;
  if (r >= NPOS_) return;
  int l = label[r];
  const float* p = breg + (size_t)r * (4 * C_) + 4 * l;
  const float* t = tgt + (size_t)r * 4;
  float s = 0.f;
  #pragma unroll
  for (int j = 0; j < 4; ++j) {
    float d = fabsf(p[j] - t[j]);
    s += (d < 1.f) ? 0.5f * d * d : d - 0.5f;
  }
  box_rows[r] = s;
}

// ---------------------------------------------------------------------------
// Kernel 3: RoI-align (1 sample/bin) of gt_mask + BCE vs selected mask logits.
// One wave32 per ROI; per_roi[r] = mean over 784 bins.
// ---------------------------------------------------------------------------
__global__ __launch_bounds__(256) void mask_rows_kernel(
    const float* __restrict__ mask_logit, const float* __restrict__ proposal,
    const float* __restrict__ gt_mask, const int* __restrict__ matched_idx,
    const int* __restrict__ label, float* __restrict__ per_roi) {
  int wave = (blockIdx.x * blockDim.x + threadIdx.x) >> 5;
  int lane = threadIdx.x & 31;
  if (wave >= R_) return;
  int r = wave;

  float x1 = proposal[r * 4 + 0], y1 = proposal[r * 4 + 1];
  float x2 = proposal[r * 4 + 2], y2 = proposal[r * 4 + 3];
  float dx = x2 - x1, dy = y2 - y1;
  int b = matched_idx[r];
  int l = label[r];
  const float* fm  = gt_mask + (size_t)b * (H_ * W_);
  const float* sel = mask_logit + ((size_t)r * C_ + l) * MM_;

  float sum = 0.f;
  for (int i = lane; i < MM_; i += 32) {
    int iy = i / M_;
    int ix = i - iy * M_;
    float gx = x1 + (ix + 0.5f) * (1.0f / M_) * dx;
    float gy = y1 + (iy + 0.5f) * (1.0f / M_) * dy;
    bool valid = (gx > -1.0f) && (gx < (float)W_) && (gy > -1.0f) && (gy < (float)H_);
    float xc = fminf(fmaxf(gx, 0.f), (float)(W_ - 1));
    float yc = fminf(fmaxf(gy, 0.f), (float)(H_ - 1));
    int x0 = (int)floorf(xc);
    int y0 = (int)floorf(yc);
    int x1i = min(x0 + 1, W_ - 1);
    int y1i = min(y0 + 1, H_ - 1);
    float lx = xc - (float)x0;
    float ly = yc - (float)y0;
    float v00 = fm[y0 * W_ + x0], v01 = fm[y0 * W_ + x1i];
    float v10 = fm[y1i * W_ + x0], v11 = fm[y1i * W_ + x1i];
    float val = v00 * (1.f - ly) * (1.f - lx) + v01 * (1.f - ly) * lx +
                v10 * ly * (1.f - lx) + v11 * ly * lx;
    float tval = valid ? val : 0.f;
    float s = sel[i];
    float bce = fmaxf(s, 0.f) - s * tval + log1pf(expf(-fabsf(s)));
    sum += bce;
  }
  #pragma unroll
  for (int off = 16; off; off >>= 1) sum += __shfl_xor(sum, off, 32);
  if (lane == 0) per_roi[r] = sum * (1.0f / MM_);
}

// ---------------------------------------------------------------------------
// Kernel 4 (single wave32): scalar loss reductions + WMMA segment reduction.
// per_gt = Indicator[112 x 2048] (f32 0/1) x [per_roi | ones] via
// V_WMMA_F32_16X16X4_F32 (full f32, exact indicator, f32 accumulate).
//   A layout: lane L -> M = L%16, VGPR j -> K = k0 + j + 2*(L>=16)
//   B layout: VGPR j  -> K = k0 + j + 2*(L>=16), lane L -> N = L%16
//   D layout: VGPR j  -> M = j + 8*(L>=16),       lane L -> N = L%16
// N=0 column accumulates seg_sum, N=1 column accumulates cnt.
// ---------------------------------------------------------------------------
__global__ __launch_bounds__(32) void final_kernel(
    const float* __restrict__ per_roi, const float* __restrict__ cls_rows,
    const float* __restrict__ box_rows, const int* __restrict__ matched_idx,
    float* __restrict__ out) {
  int lane = threadIdx.x;
  int n    = lane & 15;
  int half = lane >> 4;

  // cls_loss and box_loss (deterministic strided + shuffle reduce)
  float cs = 0.f;
  for (int i = lane; i < R_; i += 32) cs += cls_rows[i];
  #pragma unroll
  for (int off = 16; off; off >>= 1) cs += __shfl_xor(cs, off, 32);
  float bs = 0.f;
  for (int i = lane; i < NPOS_; i += 32) bs += box_rows[i];
  #pragma unroll
  for (int off = 16; off; off >>= 1) bs += __shfl_xor(bs, off, 32);
  if (lane == 0) {
    out[0] = cs * (1.0f / R_);
    out[1] = bs * (1.0f / R_);
  }

  // WMMA indicator matmul: 7 M-tiles (112 rows >= G_=100), K = 2048 in steps of 4
  v8f acc[7];
  #pragma unroll
  for (int t = 0; t < 7; ++t)
    acc[t] = (v8f){0.f, 0.f, 0.f, 0.f, 0.f, 0.f, 0.f, 0.f};

  for (int k0 = 0; k0 < R_; k0 += 4) {
    int kb = k0 + half * 2;
    int m0 = matched_idx[kb];
    int m1 = matched_idx[kb + 1];
    float p0 = per_roi[kb];
    float p1 = per_roi[kb + 1];
    v2f bfrag;
    bfrag.x = (n == 0) ? p0 : ((n == 1) ? 1.0f : 0.0f);
    bfrag.y = (n == 0) ? p1 : ((n == 1) ? 1.0f : 0.0f);
    #pragma unroll
    for (int t = 0; t < 7; ++t) {
      int g = t * 16 + n;  // A-matrix row handled by this lane
      v2f afrag;
      afrag.x = (m0 == g) ? 1.0f : 0.0f;
      afrag.y = (m1 == g) ? 1.0f : 0.0f;
      acc[t] = __builtin_amdgcn_wmma_f32_16x16x4_f32(
          /*neg_a=*/false, afrag, /*neg_b=*/false, bfrag,
          /*c_mod=*/(short)0, acc[t], /*reuse_a=*/false, /*reuse_b=*/false);
    }
  }

  // Extract: seg at N==0 (lanes 0,16), cnt at N==1 (lanes 1,17), same VGPR j.
  #pragma unroll
  for (int t = 0; t < 7; ++t) {
    #pragma unroll
    for (int j = 0; j < 8; ++j) {
      float v = acc[t][j];
      float other = __shfl_xor(v, 1, 32);  // pairs (0,1) and (16,17)
      if (n == 0) {
        int g = t * 16 + half * 8 + j;
        if (g < G_) {
          float seg = v;
          float cnt = other;
          out[2 + g] = (cnt > 0.f) ? seg / cnt : 0.f;
        }
      }
    }
  }
}

// ---------------------------------------------------------------------------
extern "C" void kernel_launch(void* const* d_in, const int* in_sizes, int n_in,
                              void* d_out, int out_size, void* d_ws, size_t ws_size,
                              hipStream_t stream) {
  const float* class_logit       = (const float*)d_in[0];
  const float* box_regression    = (const float*)d_in[1];
  const float* regression_target = (const float*)d_in[2];
  const float* mask_logit        = (const float*)d_in[3];
  const float* proposal          = (const float*)d_in[4];
  const float* gt_mask           = (const float*)d_in[5];
  const int*   matched_idx       = (const int*)d_in[6];
  const int*   label             = (const int*)d_in[7];
  float* out = (float*)d_out;

  float* ws       = (float*)d_ws;
  float* per_roi  = ws;            // R_ floats
  float* cls_rows = ws + R_;       // R_ floats
  float* box_rows = ws + 2 * R_;   // NPOS_ floats

  cls_rows_kernel<<<R_ / 8, 256, 0, stream>>>(class_logit, label, cls_rows);
  box_rows_kernel<<<(NPOS_ + 255) / 256, 256, 0, stream>>>(
      box_regression, regression_target, label, box_rows);
  mask_rows_kernel<<<R_ / 8, 256, 0, stream>>>(
      mask_logit, proposal, gt_mask, matched_idx, label, per_roi);
  final_kernel<<<1, 32, 0, stream>>>(per_roi, cls_rows, box_rows, matched_idx, out);
}